// Classifier_27934467293855
// MI455X (gfx1250) — compile-verified
//
#include <hip/hip_runtime.h>
#include <hip/hip_bf16.h>
#include <math.h>

// ---------------- problem dims (match reference) ----------------
#define B_      64
#define D_IN_   5000
#define HID_    1000
#define G_      512
#define K_      8
#define M_      32
#define GROUP_  512
#define C_      (K_ * GROUP_)       // 4096
#define NLAB_   262144

typedef float v2f __attribute__((ext_vector_type(2)));
typedef float v8f __attribute__((ext_vector_type(8)));

// =====================================================================
// Kernel 1: h = relu(x @ W1^T + b1)   [64 x 1000], fp32 WMMA 16x16x4
// One wave (32 lanes) per 16x16 output tile.
//   A (16x4, f32): VGPR0 lanes0-15 K=0 / lanes16-31 K=2; VGPR1 K=1 / K=3
//   B (4x16, f32): mirrored row striping (VGPR0 rows K=0|2, VGPR1 K=1|3)
//   C/D (16x16):   VGPR v -> M=v (lanes0-15), M=v+8 (lanes16-31), N=lane&15
// =====================================================================
__global__ __launch_bounds__(32)
void gemm1_relu_wmma(const float* __restrict__ x, const float* __restrict__ W1,
                     const float* __restrict__ b1, float* __restrict__ h)
{
    const int n0   = blockIdx.x * 16;          // column tile of h (row of W1)
    const int m0   = blockIdx.y * 16;          // row tile (sample)
    const int lane = threadIdx.x;              // 0..31 (wave32)
    const int hl   = lane >> 4;                // lane half
    const int li   = lane & 15;

    const int m = m0 + li;                     // x row for A operand
    int n = n0 + li;                           // W1 row for B operand
    if (n >= HID_) n = HID_ - 1;               // clamp (tail results discarded)

    const float* xrow = x  + (size_t)m * D_IN_;
    const float* wrow = W1 + (size_t)n * D_IN_;

    v8f c = {};
    #pragma unroll 4
    for (int k = 0; k < D_IN_; k += 4) {
        v2f a = *reinterpret_cast<const v2f*>(xrow + k + 2 * hl);  // K=2h,2h+1
        v2f b = *reinterpret_cast<const v2f*>(wrow + k + 2 * hl);
        c = __builtin_amdgcn_wmma_f32_16x16x4_f32(
                false, a, false, b, (short)0, c, false, false);
    }

    const int nc = n0 + li;
    if (nc < HID_) {
        const float bias = b1[nc];
        #pragma unroll
        for (int v = 0; v < 8; ++v) {
            const int mr = m0 + v + 8 * hl;
            h[(size_t)mr * HID_ + nc] = fmaxf(c[v] + bias, 0.0f);
        }
    }
}

// =====================================================================
// Kernel 2: y = sigmoid(h @ W2^T + b2)  [64 x 512]  + BCE(y, group_labels)
// =====================================================================
__global__ __launch_bounds__(32)
void gemm2_sig_wmma(const float* __restrict__ h, const float* __restrict__ W2,
                    const float* __restrict__ b2, const float* __restrict__ gl,
                    float* __restrict__ y, float* __restrict__ partial2)
{
    const int n0   = blockIdx.x * 16;
    const int m0   = blockIdx.y * 16;
    const int lane = threadIdx.x;
    const int hl   = lane >> 4;
    const int li   = lane & 15;

    const float* hrow = h  + (size_t)(m0 + li) * HID_;
    const float* wrow = W2 + (size_t)(n0 + li) * HID_;

    v8f c = {};
    #pragma unroll 4
    for (int k = 0; k < HID_; k += 4) {
        v2f a = *reinterpret_cast<const v2f*>(hrow + k + 2 * hl);
        v2f b = *reinterpret_cast<const v2f*>(wrow + k + 2 * hl);
        c = __builtin_amdgcn_wmma_f32_16x16x4_f32(
                false, a, false, b, (short)0, c, false, false);
    }

    const int nc = n0 + li;
    const float bias = b2[nc];
    float lsum = 0.0f;
    #pragma unroll
    for (int v = 0; v < 8; ++v) {
        const int mr = m0 + v + 8 * hl;
        const float z = 1.0f / (1.0f + __expf(-(c[v] + bias)));   // sigmoid
        y[(size_t)mr * G_ + nc] = z;
        const float t = gl[(size_t)mr * G_ + nc];
        // logaddexp(0,z) - z*t  (stable softplus)
        lsum += fmaxf(z, 0.0f) + log1pf(__expf(-fabsf(z))) - z * t;
    }
    #pragma unroll
    for (int o = 16; o > 0; o >>= 1) lsum += __shfl_xor(lsum, o, 32);
    if (lane == 0) partial2[blockIdx.y * gridDim.x + blockIdx.x] = lsum;
}

// =====================================================================
// Kernel 3: top-8 per row of y (stable: ties -> lowest index), like lax.top_k
// =====================================================================
__global__ __launch_bounds__(128)
void topk_rows(const float* __restrict__ y, int* __restrict__ idxout)
{
    __shared__ float work[G_];
    __shared__ float rv[128];
    __shared__ int   ri[128];
    const int b = blockIdx.x, tid = threadIdx.x;

    for (int j = tid; j < G_; j += 128) work[j] = y[(size_t)b * G_ + j];
    __syncthreads();

    for (int pass = 0; pass < K_; ++pass) {
        float bv = -INFINITY; int bi = 0x7fffffff;
        for (int j = tid; j < G_; j += 128) {
            const float v = work[j];
            if (v > bv || (v == bv && j < bi)) { bv = v; bi = j; }
        }
        rv[tid] = bv; ri[tid] = bi;
        __syncthreads();
        for (int s = 64; s > 0; s >>= 1) {
            if (tid < s) {
                const float v2 = rv[tid + s]; const int i2 = ri[tid + s];
                if (v2 > rv[tid] || (v2 == rv[tid] && i2 < ri[tid])) {
                    rv[tid] = v2; ri[tid] = i2;
                }
            }
            __syncthreads();
        }
        if (tid == 0) { idxout[b * K_ + pass] = ri[0]; work[ri[0]] = -INFINITY; }
        __syncthreads();
    }
}

// =====================================================================
// Kernel 4: gathered GEMV scoring + targets + BCE partials (bandwidth stage)
// One block per (b,k): 8 waves, each wave owns 64 rows of the 512-row group.
// Per lane: y chunk lives in 16 VGPRs; 512B/row fully-coalesced float4 reads.
// =====================================================================
__global__ __launch_bounds__(256)
void score_kernel(const float* __restrict__ y, const int* __restrict__ topidx,
                  const int* __restrict__ group_y, const int* __restrict__ candidates,
                  const float* __restrict__ labels, const float* __restrict__ emb,
                  float* __restrict__ logits_out, float* __restrict__ partial1)
{
    const int k = blockIdx.x;        // top-k slot 0..7
    const int b = blockIdx.y;        // sample 0..63
    const int tid  = threadIdx.x;
    const int lane = tid & 31;
    const int wv   = tid >> 5;

    __shared__ int   tgt[M_];
    __shared__ float zrow[GROUP_];
    __shared__ float wsum[8];

    const int g = topidx[b * K_ + k];

    if (tid < M_) {
        const float lab = labels[b * M_ + tid];
        tgt[tid] = (lab > 0.5f) ? candidates[b * M_ + tid] : -1;
    }
    __syncthreads();

    // register-resident y slice for this lane: y[b, 128q + 4*lane .. +3]
    float4 yv[4];
    #pragma unroll
    for (int q = 0; q < 4; ++q)
        yv[q] = *reinterpret_cast<const float4*>(y + (size_t)b * G_ + q * 128 + lane * 4);

    const int gybase = g * GROUP_;
    float lloss = 0.0f;

    for (int r = 0; r < GROUP_ / 8; ++r) {        // 64 rows per wave
        const int j   = wv * (GROUP_ / 8) + r;
        const int cid = group_y[gybase + j];       // wave-uniform -> scalar load
        const float* erow = emb + (size_t)cid * G_;

        // prefetch next row of this group (usually contiguous in emb)
        if (r + 1 < GROUP_ / 8)
            __builtin_prefetch(erow + G_ + lane * 4, 0, 0);

        float acc = 0.0f;
        #pragma unroll
        for (int q = 0; q < 4; ++q) {
            const float4 e = *reinterpret_cast<const float4*>(erow + q * 128 + lane * 4);
            acc += e.x * yv[q].x + e.y * yv[q].y + e.z * yv[q].z + e.w * yv[q].w;
        }
        #pragma unroll
        for (int o = 16; o > 0; o >>= 1) acc += __shfl_xor(acc, o, 32);

        if (lane == 0) {
            float t = 0.0f;
            #pragma unroll
            for (int m = 0; m < M_; ++m) t = (tgt[m] == cid) ? 1.0f : t;
            zrow[j] = acc;
            lloss += fmaxf(acc, 0.0f) + log1pf(__expf(-fabsf(acc))) - acc * t;
        }
    }

    if (lane == 0) wsum[wv] = lloss;
    __syncthreads();

    // coalesced logits store: d_out[b, k*512 + j]
    float* dst = logits_out + (size_t)b * C_ + (size_t)k * GROUP_;
    for (int j = tid; j < GROUP_; j += 256) dst[j] = zrow[j];

    if (tid == 0) {
        float s = 0.0f;
        #pragma unroll
        for (int w = 0; w < 8; ++w) s += wsum[w];
        partial1[b * K_ + k] = s;
    }
}

// =====================================================================
// Kernel 5: deterministic sequential reduction -> loss scalar
// =====================================================================
__global__ void finalize_loss(const float* __restrict__ partial1,
                              const float* __restrict__ partial2,
                              float* __restrict__ out_loss)
{
    if (threadIdx.x == 0 && blockIdx.x == 0) {
        float s1 = 0.0f, s2 = 0.0f;
        for (int i = 0; i < B_ * K_; ++i)        s1 += partial1[i];
        for (int i = 0; i < (B_/16)*(G_/16); ++i) s2 += partial2[i];
        *out_loss = s1 / (float)(B_ * C_) + s2 / (float)(B_ * G_);
    }
}

// =====================================================================
extern "C" void kernel_launch(void* const* d_in, const int* in_sizes, int n_in,
                              void* d_out, int out_size, void* d_ws, size_t ws_size,
                              hipStream_t stream)
{
    (void)in_sizes; (void)n_in; (void)out_size; (void)ws_size;

    const float* x      = (const float*)d_in[0];
    const int*   cand   = (const int*)  d_in[1];
    const float* labels = (const float*)d_in[2];
    const float* glab   = (const float*)d_in[3];
    const float* W1     = (const float*)d_in[4];
    const float* b1     = (const float*)d_in[5];
    const float* W2     = (const float*)d_in[6];
    const float* b2     = (const float*)d_in[7];
    const float* emb    = (const float*)d_in[8];
    const int*   gy     = (const int*)  d_in[9];

    float* logits = (float*)d_out;               // [64, 4096]
    float* loss   = logits + (size_t)B_ * C_;    // scalar at the end

    // workspace carve-up (16B aligned)
    char* ws = (char*)d_ws;
    float* h        = (float*)(ws);                        // 64*1000 f32 = 256000 B
    float* y        = (float*)(ws + 256000);               // 64*512  f32 = 131072 B
    int*   topidx   = (int*)  (ws + 387072);               // 64*8 i32    =   2048 B
    float* partial1 = (float*)(ws + 389120);               // 512 f32     =   2048 B
    float* partial2 = (float*)(ws + 391168);               // 128 f32     =    512 B

    gemm1_relu_wmma<<<dim3((HID_ + 15) / 16, B_ / 16), 32, 0, stream>>>(x, W1, b1, h);
    gemm2_sig_wmma <<<dim3(G_ / 16, B_ / 16),          32, 0, stream>>>(h, W2, b2, glab, y, partial2);
    topk_rows      <<<B_, 128, 0, stream>>>(y, topidx);
    score_kernel   <<<dim3(K_, B_), 256, 0, stream>>>(y, topidx, gy, cand, labels,
                                                      emb, logits, partial1);
    finalize_loss  <<<1, 32, 0, stream>>>(partial1, partial2, loss);
}